// DTWMaskCalculator_23914377904869
// MI455X (gfx1250) — compile-verified
//
#include <hip/hip_runtime.h>

// out[i,j] = (sd < 5 ? exp(-sd^2) : 0) * (lab[i]==lab[j] ? 1 : 0.1) * exp(-dtw^2)
//
// Memory-bound stream: 768 MB @ 23.3 TB/s => ~33 us floor. No matrix structure
// worth WMMA (A_ext is one v_cmp/v_cndmask per element). This version moves the
// two streamed inputs through the CDNA5 async global->LDS path
// (GLOBAL_LOAD_ASYNC_TO_LDS_B128, ASYNCcnt, s_wait_asynccnt) with per-thread
// double buffering: next chunk is in flight while the current one is consumed
// from LDS. Async-load completions retire in order, so s_wait_asynccnt 2 with
// 4 outstanding guarantees the current buffer's pair has landed -- no barriers
// needed since each thread reads only its own 16B slots.

typedef float __attribute__((ext_vector_type(4))) f32x4;
typedef int   __attribute__((ext_vector_type(4))) i32x4;

#define THREADS 256

__device__ __forceinline__ void async_load_b128(unsigned lds_off, const void* gptr) {
    // VDST = LDS byte address (VGPR), VADDR = 64-bit global address (VGPR pair)
    asm volatile("global_load_async_to_lds_b128 %0, %1, off"
                 :: "v"(lds_off), "v"((unsigned long long)(uintptr_t)gptr)
                 : "memory");
}

__global__ __launch_bounds__(THREADS) void dtw_mask_kernel_async(
    const float* __restrict__ sd,
    const float* __restrict__ dtw,
    const int*   __restrict__ labels,
    float*       __restrict__ out,
    long long nvec4,      // number of float4 chunks (N*N/4)
    int nmask,            // N-1  (N is a power of two)
    int nshift)           // log2(N)
{
    // [buffer][array: 0=sd 1=dtw][thread] -> 16 KB total, 16B-aligned slots
    __shared__ f32x4 sbuf[2][2][THREADS];

    const unsigned  t      = threadIdx.x;
    const long long stride = (long long)gridDim.x * THREADS;
    long long cur          = (long long)blockIdx.x * THREADS + t;

    // 32-bit LDS byte offsets of this thread's slots (flat LDS addr bits[31:0])
    const unsigned lds_s0 = (unsigned)(uintptr_t)&sbuf[0][0][t];
    const unsigned lds_t0 = (unsigned)(uintptr_t)&sbuf[0][1][t];
    const unsigned lds_s1 = (unsigned)(uintptr_t)&sbuf[1][0][t];
    const unsigned lds_t1 = (unsigned)(uintptr_t)&sbuf[1][1][t];

    // Prologue: stage iteration 0 into buffer 0.
    if (cur < nvec4) {
        async_load_b128(lds_s0, (const char*)sd  + (cur << 4));
        async_load_b128(lds_t0, (const char*)dtw + (cur << 4));
    }

    int buf = 0;
    for (; cur < nvec4; cur += stride) {
        const long long nxt = cur + stride;
        if (nxt < nvec4) {
            // Prefetch next chunk into the other buffer, then wait for the
            // current pair: in-order async completion => ASYNCcnt<=2 is enough.
            async_load_b128(buf ? lds_s0 : lds_s1, (const char*)sd  + (nxt << 4));
            async_load_b128(buf ? lds_t0 : lds_t1, (const char*)dtw + (nxt << 4));
            asm volatile("s_wait_asynccnt 0x2" ::: "memory");
        } else {
            asm volatile("s_wait_asynccnt 0x0" ::: "memory");
        }

        const f32x4 s  = sbuf[buf][0][t];   // ds_load_b128
        const f32x4 td = sbuf[buf][1][t];

        const unsigned e   = (unsigned)(cur << 2);  // fits 32 bits for N=8192
        const int      row = (int)(e >> nshift);
        const int      col = (int)(e & (unsigned)nmask);

        // Labels: tiny (N ints), temporal so they stay cache-resident.
        const i32x4 lc = *(const i32x4*)(labels + col);  // 16B-aligned
        const int   lr = labels[row];

        f32x4 r;
#pragma unroll
        for (int k = 0; k < 4; ++k) {
            const float sv  = s[k];
            const float tv  = td[k];
            const float e1  = __expf(-sv * sv);          // v_exp_f32
            const float e2  = __expf(-tv * tv);
            const float ext = (lc[k] == lr) ? 1.0f : 0.1f;
            const float val = e1 * e2 * ext;
            r[k] = (sv < 5.0f) ? val : 0.0f;             // gate -> v_cndmask
        }

        __builtin_nontemporal_store(r, (f32x4*)out + cur);  // th:NT stream out

        buf ^= 1;
    }
}

extern "C" void kernel_launch(void* const* d_in, const int* in_sizes, int n_in,
                              void* d_out, int out_size, void* d_ws, size_t ws_size,
                              hipStream_t stream) {
    // setup_inputs order: adj_mx (unused), sd_mx, dtw_matrix, cluster_labels
    const float* sd     = (const float*)d_in[1];
    const float* dtw    = (const float*)d_in[2];
    const int*   labels = (const int*)d_in[3];
    float*       out    = (float*)d_out;

    const int n = in_sizes[3];                     // N (8192), power of two
    int nshift = 0;
    while ((1 << nshift) < n) ++nshift;
    const int nmask = n - 1;

    const long long nvec4 = (long long)out_size / 4;

    long long want = (nvec4 + (long long)THREADS * 4 - 1) / ((long long)THREADS * 4);
    if (want < 1) want = 1;
    if (want > (1LL << 20)) want = (1LL << 20);
    const int blocks = (int)want;                  // ~16384 for N=8192

    dtw_mask_kernel_async<<<blocks, THREADS, 0, stream>>>(sd, dtw, labels, out,
                                                          nvec4, nmask, nshift);
}